// WrapperLinear_23596550324544
// MI455X (gfx1250) — compile-verified
//
#include <hip/hip_runtime.h>

typedef __attribute__((ext_vector_type(2))) float v2f;
typedef __attribute__((ext_vector_type(8))) float v8f;

#define GROUP 32
#define BM 128
#define BN 128
#define BK 32
#define XS_STRIDE 36   // multiple of 4 -> 16B-aligned async B128 LDS dests;
                       // 36*row mod 64 distinct for 16 rows -> conflict-free A reads
#define WT_STRIDE 129  // odd stride -> conflict-free transposed DS stores

static __device__ __forceinline__ unsigned lds_off(const void* p) {
  // Flat LDS-aperture addresses carry the LDS byte offset in the low 32 bits.
  return (unsigned)(uintptr_t)p;
}

// ---------------------------------------------------------------------------
// Pre-pass: one wave32 per quantization group (lane = element within group).
// Stores an aligned quad per group: {scale, zero_point, 1/scale, scale*zp}.
// ---------------------------------------------------------------------------
__global__ __launch_bounds__(256) void fq_scales_kernel(
    const float* __restrict__ W,
    const float* __restrict__ min_s,
    const float* __restrict__ max_s,
    float4* __restrict__ sz,
    int in_f, int num_groups, long long total_groups) {
  const long long gid =
      (long long)blockIdx.x * (blockDim.x >> 5) + (threadIdx.x >> 5);
  if (gid >= total_groups) return;
  const int lane = threadIdx.x & 31;
  const long long o = gid / num_groups;
  const int g = (int)(gid % num_groups);

  float w = W[o * in_f + (long long)g * GROUP + lane];
  float wmin = w, wmax = w;
#pragma unroll
  for (int off = 16; off >= 1; off >>= 1) {
    wmin = fminf(wmin, __shfl_xor(wmin, off, 32));
    wmax = fmaxf(wmax, __shfl_xor(wmax, off, 32));
  }
  wmin *= min_s[o * num_groups + g];
  wmax *= max_s[o * num_groups + g];
  const float scale = fmaxf((wmax - wmin) * (1.0f / 255.0f), 1e-8f);
  const float zp = -wmin / scale;
  if (lane == 0) {
    float4 q;
    q.x = scale;
    q.y = zp;
    q.z = 1.0f / scale;
    q.w = scale * zp;
    sz[gid] = q;
  }
}

// ---------------------------------------------------------------------------
// Fused dequant + GEMM: D[m,n] = sum_k X[m,k] * Wdq[n,k] + bias[n]
// 128x128 tile / 256 threads, BK=32 (one quant group per k-step).
// Double-buffered LDS; x tile via async global->LDS B128 (ASYNCcnt),
// weight tile via register-staged pipeline with on-the-fly dequant.
// Per wave: 64x32 sub-tile = 4x2 f32 accumulators, v_wmma_f32_16x16x4_f32.
// ---------------------------------------------------------------------------
__global__ __launch_bounds__(256, 2) void fq_gemm_kernel(
    const float* __restrict__ X, const float* __restrict__ W,
    const float* __restrict__ V, const float4* __restrict__ SZ,
    const float* __restrict__ bias, float* __restrict__ Out,
    int M, int N, int K, int num_groups) {
  __shared__ float xs[2][BM * XS_STRIDE];
  __shared__ float wt[2][BK * WT_STRIDE];

  const int tid = threadIdx.x;
  const int lane = tid & 31;
  const int wid = tid >> 5;
  const int l15 = lane & 15;
  const int hi = lane >> 4;  // 0: lanes 0-15, 1: lanes 16-31

  const int m0 = blockIdx.y * BM;
  const int n0 = blockIdx.x * BN;
  const int mBase = (wid >> 2) * 64;  // 2 waves along M
  const int nBase = (wid & 3) * 32;   // 4 waves along N

  // Shared loader mapping: 4 consecutive k per thread, rows r+32j.
  const int lk = (tid & 7) * 4;  // k offset within BK (0..28, x4)
  const int lr = tid >> 3;       // 0..31

  v8f acc[4][2];
#pragma unroll
  for (int mt = 0; mt < 4; ++mt)
#pragma unroll
    for (int nt = 0; nt < 2; ++nt)
      acc[mt][nt] = (v8f){0.f, 0.f, 0.f, 0.f, 0.f, 0.f, 0.f, 0.f};

  float4 wreg[4], vreg[4], szreg[4];

  // Issue async global->LDS loads for the x tile at k0 into xsb.
  auto load_x_async = [&](float* xsb, int k0) {
#pragma unroll
    for (int j = 0; j < 4; ++j) {
      const int r = lr + 32 * j;
      const unsigned dst = lds_off(xsb + r * XS_STRIDE + lk);
      const float* src = X + (long long)(m0 + r) * K + k0 + lk;
      asm volatile("global_load_async_to_lds_b128 %0, %1, off"
                   :: "v"(dst), "v"(src)
                   : "memory");
    }
  };

  // Stage W/V/scale-quads for the k0 tile into registers (latency overlapped
  // with the compute that follows in program order).
  auto stage_w = [&](int k0) {
    const int g = k0 >> 5;  // one quant group per BK step
#pragma unroll
    for (int j = 0; j < 4; ++j) {
      const long long o = n0 + lr + 32 * j;  // output-feature row
      wreg[j] = *(const float4*)(W + o * K + k0 + lk);
      vreg[j] = *(const float4*)(V + o * K + k0 + lk);
      szreg[j] = SZ[o * num_groups + g];
    }
  };

  // Dequantize staged registers and store transposed [k][n] into wtb.
  auto dequant_store = [&](float* wtb) {
#pragma unroll
    for (int j = 0; j < 4; ++j) {
      const int r = lr + 32 * j;  // column n within tile
      const float s = szreg[j].x, zp = szreg[j].y;
      const float inv_s = szreg[j].z, szp = szreg[j].w;
      const float wv[4] = {wreg[j].x, wreg[j].y, wreg[j].z, wreg[j].w};
      const float vv[4] = {vreg[j].x, vreg[j].y, vreg[j].z, vreg[j].w};
#pragma unroll
      for (int i = 0; i < 4; ++i) {
        const float q =
            fminf(fmaxf(rintf(fmaf(wv[i], inv_s, zp + vv[i])), 0.0f), 255.0f);
        wtb[(lk + i) * WT_STRIDE + r] = fmaf(q, s, -szp);
      }
    }
  };

  // 64 WMMAs on the current buffers.
  auto compute = [&](const float* xsb, const float* wtb) {
#pragma unroll
    for (int kb = 0; kb < BK; kb += 4) {
      const int ka = kb + hi * 2;  // A/B vgpr0 holds K = ka (ISA layout)
      v2f a[4], b[2];
#pragma unroll
      for (int mt = 0; mt < 4; ++mt) {
        const float* p = xsb + (mBase + mt * 16 + l15) * XS_STRIDE + ka;
        a[mt] = (v2f){p[0], p[1]};
      }
#pragma unroll
      for (int nt = 0; nt < 2; ++nt) {
        const int nn = nBase + nt * 16 + l15;
        b[nt] = (v2f){wtb[(ka + 0) * WT_STRIDE + nn],
                      wtb[(ka + 1) * WT_STRIDE + nn]};
      }
#pragma unroll
      for (int mt = 0; mt < 4; ++mt)
#pragma unroll
        for (int nt = 0; nt < 2; ++nt)
          acc[mt][nt] = __builtin_amdgcn_wmma_f32_16x16x4_f32(
              false, a[mt], false, b[nt], (short)0, acc[mt][nt], false, false);
    }
  };

  // ---- pipeline prologue: fill buffer 0 ----
  load_x_async(xs[0], 0);
  stage_w(0);
  dequant_store(wt[0]);
  asm volatile("s_wait_asynccnt 0x0" ::: "memory");
  __syncthreads();

  const int KIT = K / BK;
  for (int it = 0; it < KIT; ++it) {
    const int cur = it & 1;
    const int nxt = cur ^ 1;
    const bool more = (it + 1) < KIT;
    if (more) {
      load_x_async(xs[nxt], (it + 1) * BK);  // async, overlaps compute
      stage_w((it + 1) * BK);                // global loads in flight
    }
    compute(xs[cur], wt[cur]);
    if (more) dequant_store(wt[nxt]);        // consume staged regs
    asm volatile("s_wait_asynccnt 0x0" ::: "memory");
    __syncthreads();
  }

  // ---- epilogue: add bias, store (lanes consecutive in n -> coalesced) ----
#pragma unroll
  for (int mt = 0; mt < 4; ++mt) {
#pragma unroll
    for (int nt = 0; nt < 2; ++nt) {
      const int n = n0 + nBase + nt * 16 + l15;
      const float bv = bias[n];
#pragma unroll
      for (int r = 0; r < 8; ++r) {
        const int m = m0 + mBase + mt * 16 + r + hi * 8;
        Out[(long long)m * N + n] = acc[mt][nt][r] + bv;
      }
    }
  }
}

// ---------------------------------------------------------------------------
extern "C" void kernel_launch(void* const* d_in, const int* in_sizes, int n_in,
                              void* d_out, int out_size, void* d_ws,
                              size_t ws_size, hipStream_t stream) {
  const float* x = (const float*)d_in[0];
  const float* w = (const float*)d_in[1];
  const float* val = (const float*)d_in[2];
  const float* mins = (const float*)d_in[3];
  const float* maxs = (const float*)d_in[4];
  const float* bias = (const float*)d_in[5];
  float* out = (float*)d_out;

  const int out_f = in_sizes[5];               // bias
  const int num_groups = in_sizes[3] / out_f;  // min_scale = [out_f, groups]
  const int in_f = num_groups * GROUP;
  const int M = in_sizes[0] / in_f;            // x = [M, in_f] flattened

  float4* sz = (float4*)d_ws;  // out_f*num_groups quads (~8 MB)

  const long long total_groups = (long long)out_f * num_groups;
  const int waves_per_block = 256 / 32;
  const long long nblk = (total_groups + waves_per_block - 1) / waves_per_block;
  fq_scales_kernel<<<dim3((unsigned)nblk), dim3(256), 0, stream>>>(
      w, mins, maxs, sz, in_f, num_groups, total_groups);

  dim3 grid(out_f / BN, M / BM);
  fq_gemm_kernel<<<grid, dim3(256), 0, stream>>>(x, w, val, sz, bias, out, M,
                                                 out_f, in_f, num_groups);
}